// correlation1D_map_V1_21801253995201
// MI455X (gfx1250) — compile-verified
//
#include <hip/hip_runtime.h>

// corr[n,i,h,w] = sum_c x[n,c,h,w] * y[n,c,h,w-i]   (0 when w-i<0)
// N=4 C=64 H=128 W=256 D=48.  Memory-bound (~88MB @ 23.3TB/s ~ 3.8us).
// Banded G = Y^T X computed with exact-fp32 V_WMMA_F32_16X16X4_F32.
// LDS holds X,Y in K-pair-interleaved layout so every WMMA operand is a
// single aligned ds_load_b64 (no VGPR pairing moves).

typedef float v2f __attribute__((ext_vector_type(2)));
typedef float v4f __attribute__((ext_vector_type(4)));
typedef float v8f __attribute__((ext_vector_type(8)));

#define N_   4
#define C_   64
#define H_   128
#define W_   256
#define D_   48
#define HW_  (H_ * W_)       // 32768, stride between c-planes
#define XPS  544             // X pair-row stride (floats): 256*2 + 32 pad -> shift 32 banks
#define YPS  608             // Y pair-row stride: (48+256)*2, 608%64==32 naturally
#define YPO  (32 * XPS)      // Y region offset = 17408 floats (16B aligned)
#define SMEMF (YPO + 32 * YPS)  // 36864 floats = 147,456 B

__global__ __launch_bounds__(512) void corr1d_wmma_f32(
    const float* __restrict__ x, const float* __restrict__ y,
    float* __restrict__ out) {
  __shared__ __align__(16) float smem[SMEMF];

  const int tid = threadIdx.x;
  const int n   = blockIdx.x >> 7;     // N*H = 512 blocks
  const int h   = blockIdx.x & 127;
  const long base = (long)n * C_ * HW_ + (long)h * W_;
  const float* xb = x + base;
  const float* yb = y + base;

  // ---- stage X[64][256] pair-interleaved: (c,w) -> smem[(c>>1)*XPS + w*2 + (c&1)]
#pragma unroll
  for (int it = 0; it < 4; ++it) {               // 32 pairs * 64 w-quads = 2048
    int idx = tid + it * 512;
    int c2 = idx >> 6, q = idx & 63;
    const v4f* r0 = (const v4f*)(xb + (long)(2 * c2) * HW_);
    const v4f* r1 = (const v4f*)(xb + (long)(2 * c2 + 1) * HW_);
    v4f a = __builtin_nontemporal_load(r0 + q);
    v4f b = __builtin_nontemporal_load(r1 + q);
    v4f t0 = {a.x, b.x, a.y, b.y};
    v4f t1 = {a.z, b.z, a.w, b.w};
    float* dst = &smem[c2 * XPS + 8 * q];
    ((v4f*)dst)[0] = t0;
    ((v4f*)dst)[1] = t1;
  }

  // ---- stage Y with 48-element zero left pad, same interleave ----
  {
    v4f zz = {0.f, 0.f, 0.f, 0.f};
#pragma unroll
    for (int it = 0; it < 2; ++it) {             // 32 pairs * 24 v4f of pad = 768
      int idx = tid + it * 512;
      if (idx < 32 * 24) {
        int c2 = idx / 24, q = idx % 24;
        *(v4f*)&smem[YPO + c2 * YPS + 4 * q] = zz;
      }
    }
#pragma unroll
    for (int it = 0; it < 4; ++it) {             // 2048 interleave tasks
      int idx = tid + it * 512;
      int c2 = idx >> 6, q = idx & 63;
      const v4f* r0 = (const v4f*)(yb + (long)(2 * c2) * HW_);
      const v4f* r1 = (const v4f*)(yb + (long)(2 * c2 + 1) * HW_);
      v4f a = __builtin_nontemporal_load(r0 + q);
      v4f b = __builtin_nontemporal_load(r1 + q);
      v4f t0 = {a.x, b.x, a.y, b.y};
      v4f t1 = {a.z, b.z, a.w, b.w};
      float* dst = &smem[YPO + c2 * YPS + 96 + 8 * q];
      ((v4f*)dst)[0] = t0;
      ((v4f*)dst)[1] = t1;
    }
  }
  __syncthreads();

  // ---- per-wave banded GEMM: D[m=u, n=w] += Y[c,u] * X[c,w] ----
  const int wave = tid >> 5;
  const int lane = tid & 31;
  const int ln   = lane & 15;          // M/N lane index
  const int hg   = lane >> 4;          // half-group selects K pair
  const int w0   = wave * 16;          // this wave's w-tile

  v8f acc[4] = {};                     // u-tiles w0-48, w0-32, w0-16, w0

#pragma unroll 4
  for (int k = 0; k < C_; k += 4) {
    const int c2 = (k >> 1) + hg;      // K pair = {2*c2, 2*c2+1} = {k+2hg, k+2hg+1}
    v2f b = *(const v2f*)&smem[c2 * XPS + (w0 + ln) * 2];          // one ds_load_b64
#pragma unroll
    for (int t = 0; t < 4; ++t) {
      v2f a = *(const v2f*)&smem[YPO + c2 * YPS + (w0 + 16 * t + ln) * 2];
      acc[t] = __builtin_amdgcn_wmma_f32_16x16x4_f32(
          false, a, false, b, (short)0, acc[t], false, false);
    }
  }

  // ---- extract valid diagonals i = w-u into LDS [48][256] (aliases X region) ----
  __syncthreads();                     // staging fully consumed
#pragma unroll
  for (int t = 0; t < 4; ++t) {
#pragma unroll
    for (int r = 0; r < 8; ++r) {
      int m = r + 8 * hg;              // row of D tile
      int i = 48 - 16 * t + ln - m;    // disparity = w - u
      if (i >= 0 && i < D_) smem[i * W_ + w0 + ln] = acc[t][r];
    }
  }
  __syncthreads();

  // ---- coalesced NT store: 48*256 floats = 3072 v4f ----
#pragma unroll
  for (int it = 0; it < 6; ++it) {
    int idx = tid + it * 512;
    int i = idx >> 6, q = idx & 63;
    v4f v = *(const v4f*)&smem[idx * 4];
    __builtin_nontemporal_store(
        v, (v4f*)(out + ((long)(n * D_ + i) * H_ + h) * W_) + q);
  }
}

extern "C" void kernel_launch(void* const* d_in, const int* in_sizes, int n_in,
                              void* d_out, int out_size, void* d_ws, size_t ws_size,
                              hipStream_t stream) {
  const float* x = (const float*)d_in[0];
  const float* y = (const float*)d_in[1];
  // d_in[2] = maxdisp (48), fixed by the reference; compiled in as D_.
  float* out = (float*)d_out;
  corr1d_wmma_f32<<<dim3(N_ * H_), dim3(512), 0, stream>>>(x, y, out);
}